// HierarchicalHashEmbedding_3401614098798
// MI455X (gfx1250) — compile-verified
//
#include <hip/hip_runtime.h>
#include <hip/hip_bf16.h>

// Embedding gather: out[b,l,:] = table[indices[b,l], :]
//   indices: [4096*200] int32   (d_in[0])
//   table:   [1'000'000 * 64] f32 (d_in[1])
//   out:     [4096*200*64] f32
//
// MI455X plan: pure memory-roofline kernel (zero FLOPs).
//  - One 256 B table row == 16 x b128. 16 consecutive lanes handle one row,
//    one float4 each -> coalesced global_load_b128 reads of whole rows and a
//    perfectly contiguous global_store_b128 output stream.
//  - Table loads use default RT temporal hint: the randomly-touched working
//    set (~143 MB of the 256 MB table) fits the 192 MB global L2, so repeats
//    hit L2 instead of HBM.
//  - Output (210 MB, write-once) uses non-temporal stores (TH=NT) so the
//    streaming writes don't evict the L2-resident table rows.

typedef __attribute__((ext_vector_type(4))) float v4f;

__global__ __launch_bounds__(256) void hash_embed_gather_kernel(
    const int* __restrict__ indices,   // [n_rows]
    const float* __restrict__ table,   // [VOCAB * 64]
    float* __restrict__ out,           // [n_rows * 64]
    int n_chunks)                      // n_rows * 16 (float4 chunks)
{
    int gid = blockIdx.x * blockDim.x + threadIdx.x;
    if (gid >= n_chunks) return;

    int row   = gid >> 4;   // which lookup (16 lanes share one row)
    int chunk = gid & 15;   // which float4 of the 64-float row

    // 16 adjacent lanes read the same index word -> coalesces to one request.
    int idx = indices[row];

    // Row byte offset = idx * 256; keep 64-bit math for the 256 MB table.
    const v4f* src = reinterpret_cast<const v4f*>(table)
                   + ((size_t)(unsigned)idx << 4) + chunk;
    v4f v = *src;  // RT load: keep hot table rows resident in 192 MB L2

    // NT store: stream the 210 MB output past L2 without polluting it.
    __builtin_nontemporal_store(v, reinterpret_cast<v4f*>(out) + gid);
}

extern "C" void kernel_launch(void* const* d_in, const int* in_sizes, int n_in,
                              void* d_out, int out_size, void* d_ws, size_t ws_size,
                              hipStream_t stream) {
    const int*   indices = (const int*)d_in[0];    // [4096*200] int32
    const float* table   = (const float*)d_in[1];  // [1e6 * 64] f32
    float*       out     = (float*)d_out;          // [4096*200*64] f32

    // out_size = n_rows * 64 floats; each thread moves one float4.
    int n_chunks = out_size / 4;

    const int block = 256;                         // 8 wave32 waves per block
    int grid = (n_chunks + block - 1) / block;

    hash_embed_gather_kernel<<<grid, block, 0, stream>>>(indices, table, out, n_chunks);
}